// qcir_34497177321624
// MI455X (gfx1250) — compile-verified
//
#include <hip/hip_runtime.h>

// 16-qubit batched statevector simulator for MI455X (gfx1250, wave32).
// State kept as separate real/imag fp32 planes in d_ws (L2-resident: 128 MB < 192 MB L2).
// Each YZY layer is fused into four 16x16 complex matrices (one per 4-qubit group)
// and applied as complex GEMMs built from V_WMMA_F32_16X16X4_F32.
// All state addressing uses 32-bit element offsets so the backend can emit
// SGPR-base + 32-bit-VGPR-offset global loads/stores (no v_add_nc_u64 chains).

typedef float v2f __attribute__((ext_vector_type(2)));
typedef float v8f __attribute__((ext_vector_type(8)));

#define NQ      16
#define NSTATE  65536u  // 2^16 amplitudes
#define BATCH   256

// ---------------------------------------------------------------------------
// Build the 5 layers x 4 hextet 16x16 complex matrices from the weights.
// mats layout per (L,h), stride 768 floats:
//   [+0..255]   real  (row-major a*16+b)
//   [+256..511] imag
//   [+512..767] negated imag (pre-negated so yzy_pass loads it directly)
// Per-qubit gate G = Ry(p2) * Rz(p1) * Ry(p0).
// ---------------------------------------------------------------------------
__global__ void build_mats(const float* __restrict__ weight, float* __restrict__ mats) {
    int L = blockIdx.x;            // layer 0..4
    int h = blockIdx.y;            // hextet 0..3 (wires 4h..4h+3, 4h most significant)
    int t = threadIdx.x;           // 0..255 -> matrix entry
    int a = t >> 4, b = t & 15;

    float pr = 1.f, pi = 0.f;
    #pragma unroll
    for (int j = 0; j < 4; ++j) {
        int q = 4 * h + j;
        float p0 = weight[L * 48 + q * 3 + 0];
        float p1 = weight[L * 48 + q * 3 + 1];
        float p2 = weight[L * 48 + q * 3 + 2];
        float ca = cosf(0.5f * p0), sa = sinf(0.5f * p0);
        float cb = cosf(0.5f * p1), sb = sinf(0.5f * p1);
        float cc = cosf(0.5f * p2), sc = sinf(0.5f * p2);
        float u  = cc * ca - sc * sa;
        float v  = cc * ca + sc * sa;
        float w2 = cc * sa + sc * ca;
        float x2 = cc * sa - sc * ca;
        int ab = (a >> (3 - j)) & 1;
        int bb = (b >> (3 - j)) & 1;
        float gr, gi;
        if (!ab && !bb)      { gr =  cb * u;  gi = -sb * v;  }  // G00
        else if (!ab &&  bb) { gr = -cb * w2; gi =  sb * x2; }  // G01
        else if ( ab && !bb) { gr =  cb * w2; gi =  sb * x2; }  // G10
        else                 { gr =  cb * u;  gi =  sb * v;  }  // G11
        float nr = pr * gr - pi * gi;
        float ni = pr * gi + pi * gr;
        pr = nr; pi = ni;
    }
    int base = (L * 4 + h) * 768;
    mats[base + t]       = pr;
    mats[base + 256 + t] = pi;
    mats[base + 512 + t] = -pi;
}

// ---------------------------------------------------------------------------
// Initial encoded state: product state  amp[k] = prod_w (bit_w ? sin : cos)(x_w/2).
// grid (curB, 64), block 256, 4 amplitudes per thread.
// ---------------------------------------------------------------------------
__global__ void init_state(const float* __restrict__ inputs,
                           float* __restrict__ SR, float* __restrict__ SI,
                           int chunkStart) {
    unsigned cb = blockIdx.x;
    unsigned b  = chunkStart + cb;
    __shared__ float cs0[16], cs1[16];
    unsigned t = threadIdx.x;
    if (t < 16) {
        float x = 0.5f * inputs[b * NQ + t];
        cs0[t] = cosf(x);
        cs1[t] = sinf(x);
    }
    __syncthreads();
    unsigned sbase = cb * NSTATE;
    unsigned k0 = (blockIdx.y * 256u + t) * 4u;
    #pragma unroll
    for (unsigned e = 0; e < 4; ++e) {
        unsigned k = k0 + e;
        float amp = 1.f;
        #pragma unroll
        for (int w = 0; w < 16; ++w) {
            amp *= ((k >> (15 - w)) & 1u) ? cs1[w] : cs0[w];
        }
        SR[sbase + k] = amp;
        SI[sbase + k] = 0.f;
    }
}

// ---------------------------------------------------------------------------
// Apply one 16x16 complex matrix along one hextet axis of the state tensor
// S[16][16][16][16] (in-place).  State viewed as (outer, 16, inner) with
// inner = 1<<log2inner.  Each wave computes one 16x16 tile:
//   Dr = Mr*Xr + (-Mi)*Xi ,  Di = Mr*Xi + Mi*Xr
// via 16 V_WMMA_F32_16X16X4_F32 (K accumulated in 4 chunks of 4).
// grid (curB, 32), block 256 (= 8 waves, 128 columns per block).
// Fragment layouts per CDNA5 ISA:
//   A 16x4:  lanes 0-15 row M=lane, VGPR r holds K = r + 2*(lane>=16)
//   B 4x16:  mirrored; C/D: VGPR v holds M = v + 8*(lane>=16), N = lane&15
// ---------------------------------------------------------------------------
__global__ __launch_bounds__(256)
void yzy_pass(float* __restrict__ SR, float* __restrict__ SI,
              const float* __restrict__ matR, int log2inner) {
    unsigned lane = threadIdx.x & 31u;
    unsigned wave = threadIdx.x >> 5;
    unsigned hf   = lane >> 4;          // half-wave 0/1
    unsigned l16  = lane & 15u;
    unsigned inner = 1u << log2inner;

    const float* matI = matR + 256;
    const float* matN = matR + 512;     // pre-negated imag

    v2f ar[4], ai[4], an[4];
    #pragma unroll
    for (unsigned c = 0; c < 4; ++c) {
        #pragma unroll
        for (unsigned r = 0; r < 2; ++r) {
            unsigned K = 4u * c + r + 2u * hf;
            ar[c][r] = matR[l16 * 16u + K];
            ai[c][r] = matI[l16 * 16u + K];
            an[c][r] = matN[l16 * 16u + K];
        }
    }

    unsigned sbase = blockIdx.x * NSTATE;
    unsigned col = (blockIdx.y * 8u + wave) * 16u + l16;  // column index 0..4095
    unsigned o   = col >> log2inner;
    unsigned in  = col & (inner - 1u);
    unsigned vbase = sbase + o * (inner << 4) + in;       // + K*inner selects row

    v8f dr = {0.f, 0.f, 0.f, 0.f, 0.f, 0.f, 0.f, 0.f};
    v8f di = {0.f, 0.f, 0.f, 0.f, 0.f, 0.f, 0.f, 0.f};

    #pragma unroll
    for (unsigned c = 0; c < 4; ++c) {
        v2f br, bi;
        #pragma unroll
        for (unsigned r = 0; r < 2; ++r) {
            unsigned K = 4u * c + r + 2u * hf;
            unsigned addr = vbase + K * inner;
            br[r] = SR[addr];
            bi[r] = SI[addr];
        }
        dr = __builtin_amdgcn_wmma_f32_16x16x4_f32(false, ar[c], false, br, (short)0, dr, false, false);
        dr = __builtin_amdgcn_wmma_f32_16x16x4_f32(false, an[c], false, bi, (short)0, dr, false, false);
        di = __builtin_amdgcn_wmma_f32_16x16x4_f32(false, ar[c], false, bi, (short)0, di, false, false);
        di = __builtin_amdgcn_wmma_f32_16x16x4_f32(false, ai[c], false, br, (short)0, di, false, false);
    }

    // D fragment: VGPR v holds M = v + 8*hf, same column N = l16.
    #pragma unroll
    for (unsigned v = 0; v < 8; ++v) {
        unsigned M = v + 8u * hf;
        unsigned addr = vbase + M * inner;
        SR[addr] = dr[v];
        SI[addr] = di[v];
    }
}

// ---------------------------------------------------------------------------
// In-place CNOT: swap amplitudes with (control=1, target=0) <-> (control=1, target=1).
// pc/pt are LSB bit positions.  grid (curB, 64), block 256 -> 16384 pairs.
// ---------------------------------------------------------------------------
__global__ void cnot_swap(float* __restrict__ SR, float* __restrict__ SI,
                          int pc, int pt) {
    unsigned sbase = blockIdx.x * NSTATE;
    unsigned tid = blockIdx.y * 256u + threadIdx.x;  // 0..16383
    int plo = pc < pt ? pc : pt;
    int phi = pc < pt ? pt : pc;
    unsigned mlo = (1u << plo) - 1u;
    unsigned mhi = (1u << phi) - 1u;
    unsigned k = ((tid & ~mlo) << 1) | (tid & mlo);  // insert 0 at plo
    k = ((k & ~mhi) << 1) | (k & mhi);               // insert 0 at phi
    k |= (1u << pc);                                 // control = 1, target = 0
    unsigned k1 = sbase + k;
    unsigned k2 = k1 | (1u << pt);                   // target = 1
    float r1 = SR[k1], r2 = SR[k2];
    SR[k1] = r2;
    SR[k2] = r1;
    float i1 = SI[k1], i2 = SI[k2];
    SI[k1] = i2;
    SI[k2] = i1;
}

// ---------------------------------------------------------------------------
// <Z_w> = 4 * sum_k (+-1)^{bit_w(k)} |amp_k|^2.  One block per batch element.
// Fixed-order LDS reduction (deterministic, no float atomics).
// ---------------------------------------------------------------------------
__global__ void measure_z(const float* __restrict__ SR, const float* __restrict__ SI,
                          float* __restrict__ out, int chunkStart) {
    unsigned cb = blockIdx.x;
    unsigned b  = chunkStart + cb;
    unsigned sbase = cb * NSTATE;
    __shared__ float sh[256 * 16];
    unsigned t = threadIdx.x;

    float acc[16];
    #pragma unroll
    for (int w = 0; w < 16; ++w) acc[w] = 0.f;
    for (unsigned k = t; k < NSTATE; k += 256u) {
        float r = SR[sbase + k], i = SI[sbase + k];
        float p = r * r + i * i;
        #pragma unroll
        for (int w = 0; w < 16; ++w)
            acc[w] += ((k >> (15 - w)) & 1u) ? -p : p;
    }
    #pragma unroll
    for (int w = 0; w < 16; ++w) sh[t * 16 + w] = acc[w];
    __syncthreads();
    if (t < 16) {
        float s = 0.f;
        for (int q = 0; q < 256; ++q) s += sh[q * 16 + t];
        out[b * NQ + t] = 4.f * s;
    }
}

// ---------------------------------------------------------------------------
extern "C" void kernel_launch(void* const* d_in, const int* in_sizes, int n_in,
                              void* d_out, int out_size, void* d_ws, size_t ws_size,
                              hipStream_t stream) {
    const float* inputs = (const float*)d_in[0];   // (256, 16)
    const float* weight = (const float*)d_in[1];   // (240,)
    float* out = (float*)d_out;                    // (256, 16)

    float* mats = (float*)d_ws;                    // 5*4*768 floats = 60 KB
    const size_t matsBytes = 65536;                // padded region for matrices
    const size_t perBatchBytes = (size_t)NSTATE * 2 * sizeof(float);  // 512 KB

    size_t avail = ws_size > matsBytes ? ws_size - matsBytes : 0;
    int chunkB = (int)(avail / perBatchBytes);
    if (chunkB > BATCH) chunkB = BATCH;
    if (chunkB < 1) chunkB = 1;

    build_mats<<<dim3(5, 4), 256, 0, stream>>>(weight, mats);

    float* SR = (float*)((char*)d_ws + matsBytes);
    float* SI = SR + (size_t)chunkB * NSTATE;

    for (int start = 0; start < BATCH; start += chunkB) {
        int curB = (BATCH - start < chunkB) ? (BATCH - start) : chunkB;

        init_state<<<dim3(curB, 64), 256, 0, stream>>>(inputs, SR, SI, start);

        auto run_layer = [&](int L) {
            for (int j = 0; j < 4; ++j) {
                const float* mR = mats + (size_t)(L * 4 + j) * 768;
                yzy_pass<<<dim3(curB, 32), 256, 0, stream>>>(SR, SI, mR, 4 * (3 - j));
            }
        };
        auto run_ring = [&](int step) {
            for (int i = 0; i < 16; ++i) {
                int c = i, tt = (i + step) & 15;
                cnot_swap<<<dim3(curB, 64), 256, 0, stream>>>(SR, SI, 15 - c, 15 - tt);
            }
        };

        run_layer(0); run_ring(1);
        run_layer(1); run_ring(1);
        run_layer(2); run_ring(2);
        run_layer(3); run_ring(2);
        run_layer(4);

        measure_z<<<curB, 256, 0, stream>>>(SR, SI, out, start);
    }
}